// RotationInvariantSheafLearner_9174050144890
// MI455X (gfx1250) — compile-verified
//
#include <hip/hip_runtime.h>
#include <math.h>

// CDNA5 (gfx1250) fused kernel for the sheaf-learner forward pass.
// Exact-precision path: Gram matrices + Maps combine in f32 VALU,
// final 16x16 linear layer via chained V_WMMA_F32_16X16X4_F32 (full f32 WMMA).
// Epilogue: branch-free fast tanh (v_exp_f32 + v_rcp_f32), hoisted bounds
// guard, non-temporal streaming of Maps/out to keep x resident in L2.

typedef __attribute__((ext_vector_type(2))) float v2f;   // A/B frag of 16x16x4 f32 WMMA
typedef __attribute__((ext_vector_type(4))) float v4f;   // NT 128-bit loads
typedef __attribute__((ext_vector_type(8))) float v8f;   // C/D frag (16x16 f32)

#define SH 32   // hidden channels H
// D (stalk dim) = 4, D*D = 16 features

__device__ __forceinline__ float dot4(float4 a, float4 b) {
  return a.x * b.x + a.y * b.y + a.z * b.z + a.w * b.w;
}

// Branch-free tanh: exp2(a * -2*log2(e)) = exp(-2a) <= 1, never overflows.
// v_rcp_f32 + mul instead of IEEE division (activation precision is ample).
__device__ __forceinline__ float fast_tanh(float v) {
  const float a = fabsf(v);
  const float e = __builtin_amdgcn_exp2f(a * -2.8853900817779268f); // exp(-2a)
  const float r = (1.0f - e) * __builtin_amdgcn_rcpf(1.0f + e);    // tanh(|v|)
  return copysignf(r, v);
}

__global__ __launch_bounds__(64) void sheaf_fused_kernel(
    const float* __restrict__ x,          // (N, 4, 32)
    const float* __restrict__ Maps,       // (E, 4, 4)
    const float* __restrict__ W1,         // (16, 16)
    const float* __restrict__ b1,         // (16,)
    const int*   __restrict__ edge_index, // (2, E)
    const int*   __restrict__ opp_idx,    // (E,)
    float* __restrict__ out,              // (E, 4, 4)
    int E)
{
  // Two waves per block; each wave owns 32 edges; stride-17 rows avoid LDS
  // bank conflicts (17 coprime with 64 banks).
  __shared__ float lds_sims[2][32 * 17];

  const int lane = threadIdx.x & 31;
  const int wid  = threadIdx.x >> 5;
  const long base = ((long)blockIdx.x * 2 + wid) * 32;  // first edge of wave

  // ---- Phase 1: per-lane edge -> sims[16] in LDS (f32 VALU, gathers via L2)
  const long e  = base + lane;
  const long ec = (e < E) ? e : (long)(E - 1);          // clamp, no divergence
  const int s = edge_index[ec];
  const int t = edge_index[(long)E + ec];
  const int o = opp_idx[ec];

  const float* xs = x + (long)s * (4 * SH);
  const float* xt = x + (long)t * (4 * SH);

  float Gss[4][4], Gts[4][4];
  #pragma unroll
  for (int a = 0; a < 4; ++a)
    #pragma unroll
    for (int b = 0; b < 4; ++b) { Gss[a][b] = 0.f; Gts[a][b] = 0.f; }

  #pragma unroll
  for (int hc = 0; hc < 8; ++hc) {            // H=32 in float4 chunks
    float4 A4[4], B4[4];
    #pragma unroll
    for (int r = 0; r < 4; ++r) {
      A4[r] = *(const float4*)(xs + r * SH + hc * 4);
      B4[r] = *(const float4*)(xt + r * SH + hc * 4);
    }
    #pragma unroll
    for (int a = 0; a < 4; ++a) {
      #pragma unroll
      for (int b = 0; b < 4; ++b) {
        if (b >= a) Gss[a][b] += dot4(A4[a], A4[b]);   // symmetric: upper only
        Gts[a][b] += dot4(B4[a], A4[b]);
      }
    }
  }
  Gss[1][0] = Gss[0][1]; Gss[2][0] = Gss[0][2]; Gss[3][0] = Gss[0][3];
  Gss[2][1] = Gss[1][2]; Gss[3][1] = Gss[1][3]; Gss[3][2] = Gss[2][3];

  // Maps are a read-once stream: non-temporal loads keep x resident in L2.
  const v4f* Me = (const v4f*)(Maps + (long)ec * 16);
  const v4f* Mo = (const v4f*)(Maps + (long)o  * 16);
  float* ls = &lds_sims[wid][lane * 17];
  #pragma unroll
  for (int i = 0; i < 4; ++i) {
    const v4f me = __builtin_nontemporal_load(&Me[i]);
    const v4f mo = __builtin_nontemporal_load(&Mo[i]);
    #pragma unroll
    for (int j = 0; j < 4; ++j) {
      const float v =
          me[0] * Gss[0][j] + me[1] * Gss[1][j] + me[2] * Gss[2][j] + me[3] * Gss[3][j]
        - (mo[0] * Gts[0][j] + mo[1] * Gts[1][j] + mo[2] * Gts[2][j] + mo[3] * Gts[3][j]);
      ls[4 * i + j] = 2.0f * v;               // sims flat index k = 4*i + j
    }
  }
  __syncthreads();

  // ---- Phase 2: (16 edges x 16) @ W1^T via 4 chained v_wmma_f32_16x16x4_f32
  // A layout (16x4 f32, ISA 7.12.2): v0 = K {0 | 2} per half-wave, v1 = K {1 | 3}.
  // B layout (4x16 f32, mirrored):   v0 = row K {0 | 2}, v1 = row K {1 | 3}.
  const int n     = lane & 15;                // output feature / B column
  const int hi    = (lane < 16) ? 0 : 8;      // M offset of this half-wave
  const int kband = (lane < 16) ? 0 : 2;

  v2f bfrag[4];
  #pragma unroll
  for (int c = 0; c < 4; ++c) {
    const int k0 = 4 * c + kband;
    bfrag[c].x = W1[n * 16 + k0];             // B[K][N] = W1[N][K]  (y = x W^T)
    bfrag[c].y = W1[n * 16 + k0 + 1];
  }
  const float bias = b1[n];

  // Common-case store base: out[(base + tile*16 + r + hi)*16 + n]
  float* op0 = out + base * 16 + (hi * 16 + n);
  const bool full = (base + 32) <= (long)E;   // wave-uniform

  #pragma unroll
  for (int tile = 0; tile < 2; ++tile) {
    const float* src = &lds_sims[wid][(tile * 16) * 17];
    v8f acc = {0.f, 0.f, 0.f, 0.f, 0.f, 0.f, 0.f, 0.f};
    #pragma unroll
    for (int c = 0; c < 4; ++c) {
      const int k0 = 4 * c + kband;
      v2f afrag;
      afrag.x = src[(lane & 15) * 17 + k0];
      afrag.y = src[(lane & 15) * 17 + k0 + 1];
      acc = __builtin_amdgcn_wmma_f32_16x16x4_f32(
          /*neg_a=*/false, afrag, /*neg_b=*/false, bfrag[c],
          /*c_mod=*/(short)0, acc, /*reuse_a=*/false, /*reuse_b=*/false);
    }
    // Epilogue on C/D layout: VGPR r -> edge row (r + hi), N = lane&15.
    // Output is write-once: non-temporal stores bypass L2 retention.
    if (full) {                               // uniform branch: no exec churn
      #pragma unroll
      for (int r = 0; r < 8; ++r) {
        const float y = fast_tanh(acc[r] + bias);
        __builtin_nontemporal_store(y, op0 + tile * 256 + r * 16);
      }
    } else {                                  // tail wave only
      #pragma unroll
      for (int r = 0; r < 8; ++r) {
        const long eg = base + tile * 16 + r + hi;
        if (eg < E) {
          const float y = fast_tanh(acc[r] + bias);
          __builtin_nontemporal_store(y, &out[eg * 16 + n]);
        }
      }
    }
  }
}

extern "C" void kernel_launch(void* const* d_in, const int* in_sizes, int n_in,
                              void* d_out, int out_size, void* d_ws, size_t ws_size,
                              hipStream_t stream) {
  const float* x    = (const float*)d_in[0];
  const float* Maps = (const float*)d_in[1];
  const float* W1   = (const float*)d_in[2];
  const float* b1   = (const float*)d_in[3];
  const int* edge_index = (const int*)d_in[4];
  const int* opp_idx    = (const int*)d_in[5];
  float* out = (float*)d_out;

  const int E = in_sizes[5];                      // opp_idx has E elements
  const long waves  = ((long)E + 31) / 32;        // 32 edges per wave
  const int  blocks = (int)((waves + 1) / 2);     // 2 waves per block

  sheaf_fused_kernel<<<blocks, 64, 0, stream>>>(x, Maps, W1, b1,
                                                edge_index, opp_idx, out, E);
}